// SelectMHA_80994493268159
// MI455X (gfx1250) — compile-verified
//
#include <hip/hip_runtime.h>
#include <hip/hip_bf16.h>

typedef __attribute__((ext_vector_type(2))) float v2f;
typedef __attribute__((ext_vector_type(8))) float v8f;
typedef __attribute__((ext_vector_type(4))) unsigned int u32x4;
typedef __attribute__((ext_vector_type(4))) int i32x4;
typedef __attribute__((ext_vector_type(8))) int i32x8;

constexpr int B      = 8;
constexpr int S_MAX  = 4096;
constexpr int E      = 4096;
constexpr int H      = 32;
constexpr int D      = 128;
constexpr int TOPK   = 8;
constexpr int OFFSET = 4095;           // new token position; positions 0..OFFSET valid
constexpr int QKV_ROWS = 3 * E;        // 12288
constexpr int KT     = 64;             // TDM tile rows (K/V positions per tile)
constexpr int ROWF   = D + 4;          // padded LDS row stride (floats): TDM pads 4 dw/128 dw

// ---------------------------------------------------------------------------
// WMMA GEMV inner loop: D[m][n] += sum_k W[f0+m][k] * X[n][k]
// A (16x4)=weight rows, B (4x16)=activations^T, K stepped by 4
// (v_wmma_f32_16x16x4_f32). unroll 16 keeps ~32 b64 loads in flight per wave.
// ---------------------------------------------------------------------------
__device__ __forceinline__ v8f wmma_gemv_range(const float* __restrict__ wrow,
                                               const float* __restrict__ xrow,
                                               int k0, int k1, int koff, v8f acc)
{
#pragma unroll 16
    for (int k = k0; k < k1; k += 4) {
        const float2 av = *reinterpret_cast<const float2*>(wrow + k + koff);
        const float2 bv = *reinterpret_cast<const float2*>(xrow + k + koff);
        v2f a; a[0] = av.x; a[1] = av.y;
        v2f b; b[0] = bv.x; b[1] = bv.y;
        acc = __builtin_amdgcn_wmma_f32_16x16x4_f32(
                  false, a, false, b, (short)0, acc, false, false);
    }
    return acc;
}

__device__ __forceinline__ unsigned selected_head_union(const int* __restrict__ head_idx)
{
    unsigned mask = 0u;
#pragma unroll
    for (int i = 0; i < B * TOPK; ++i) mask |= 1u << (head_idx[i] & (H - 1));
    return mask;
}

// ---------------------------------------------------------------------------
// TDM: async 2D tile load Global->LDS (D# per ISA cdna5 ch.8).
//   tile: tile_rows x 128 f32, row stride in memory = 2*H*D floats,
//   LDS padding 4 dwords every 128 dwords -> 132-float padded rows (kills the
//   32-way bank conflict when 32 lanes read 32 different rows at equal d).
//   rows beyond valid_rows read as zero (OOB rule) - used to blank the stale
//   cache row at s==OFFSET, which is then overwritten from the qkv workspace.
// Builtin arity differs per toolchain (bridge doc): clang-22 = 5 args,
// clang-23 = 6 args.
// ---------------------------------------------------------------------------
__device__ __forceinline__ void tdm_load_2d(unsigned lds_byte_addr,
                                            const float* gsrc,
                                            int valid_rows, int tile_rows)
{
    u32x4 g0;
    g0[0] = 1u;                                        // count=1, user desc
    g0[1] = lds_byte_addr;                             // lds_addr
    const unsigned long long ga = (unsigned long long)(const void*)gsrc;
    g0[2] = (unsigned)(ga & 0xffffffffull);            // global_addr[31:0]
    g0[3] = (unsigned)((ga >> 32) & 0x01ffffffull)     // global_addr[56:32]
          | (2u << 30);                                // type=2 ("image")
    i32x8 g1;
    g1[0] = (2 << 16)                                  // data_size = 4B
          | (1 << 20)                                  // pad_enable
          | (6 << 22)                                  // pad_interval: 128 dw
          | (3 << 25);                                 // pad_amount: 4 dw
    g1[1] = (D & 0xffff) << 16;                        // tensor_dim0 = 128
    g1[2] = ((valid_rows & 0xffff) << 16);             // tensor_dim1 lo
    g1[3] = ((valid_rows >> 16) & 0xffff)              // tensor_dim1 hi
          | (D << 16);                                 // tile_dim0 = 128
    g1[4] = (tile_rows & 0xffff);                      // tile_dim1; tile_dim2=0
    g1[5] = 2 * H * D;                                 // tensor_dim0_stride = 8192
    g1[6] = 0;
    g1[7] = 0;
    const i32x4 z4 = {0, 0, 0, 0};
#if defined(__clang_major__) && (__clang_major__ >= 23)
    const i32x8 z8 = {0, 0, 0, 0, 0, 0, 0, 0};
    __builtin_amdgcn_tensor_load_to_lds(g0, g1, z4, z4, z8, 0);
#else
    __builtin_amdgcn_tensor_load_to_lds(g0, g1, z4, z4, 0);
#endif
}

// ---------------------------------------------------------------------------
// Split-K GEMV partial kernel (blockIdx.y = k-slice), un-biased partials.
// ROWMASK (qkv): skip row tiles of unselected heads. !ROWMASK (out-proj):
// skip k head-chunks not in the union. Skips are wave-uniform -> EXEC full.
// ---------------------------------------------------------------------------
template <bool ROWMASK>
__global__ void __launch_bounds__(256) gemv_part_kernel(
    const float* __restrict__ W, const float* __restrict__ X,
    const int* __restrict__ head_idx, float* __restrict__ part,
    int rows, int nsplit)
{
    const unsigned mask = selected_head_union(head_idx);

    const int wave  = threadIdx.x >> 5;
    const int lane  = threadIdx.x & 31;
    const int f0    = (blockIdx.x * 8 + wave) * 16;
    const int slice = blockIdx.y;
    const int klen  = E / nsplit;
    const int k_lo  = slice * klen;
    const int k_hi  = k_lo + klen;

    if (ROWMASK) {
        const int h = (f0 % E) / D;
        if (!((mask >> h) & 1u)) return;
    }

    const int  m    = lane & 15;
    const bool hi   = lane >= 16;
    const int  koff = hi ? 2 : 0;
    const float* wrow = W + (size_t)(f0 + m) * E;
    const int  n    = lane & 15;
    const float* xrow = X + (size_t)((n < B) ? n : 0) * E;

    v8f acc = {0.f, 0.f, 0.f, 0.f, 0.f, 0.f, 0.f, 0.f};
    if (ROWMASK) {
        acc = wmma_gemv_range(wrow, xrow, k_lo, k_hi, koff, acc);
    } else {
        for (int hc = k_lo / D; hc < k_hi / D; ++hc) {
            if (!((mask >> hc) & 1u)) continue;
            acc = wmma_gemv_range(wrow, xrow, hc * D, hc * D + D, koff, acc);
        }
    }

    if (n < B) {
        float* p = part + (size_t)slice * B * rows;
#pragma unroll
        for (int r = 0; r < 8; ++r) {
            const int f = f0 + r + (hi ? 8 : 0);
            p[(size_t)n * rows + f] = acc[r];
        }
    }
}

__global__ void __launch_bounds__(256) gemv_combine_kernel(
    const float* __restrict__ part, const float* __restrict__ bias,
    float* __restrict__ out, int rows, int nsplit)
{
    const int idx = blockIdx.x * 256 + threadIdx.x;
    if (idx >= B * rows) return;
    const int f = idx % rows;
    float s = bias[f];
    for (int i = 0; i < nsplit; ++i)
        s += part[(size_t)i * B * rows + idx];
    out[idx] = s;
}

// ---------------------------------------------------------------------------
// Flash-decoding attention partial with TDM-staged K/V tiles.
// block = (b, topk slot j, s-chunk). Wave 0 drives the TDM (double-buffered,
// s_wait_tensorcnt), all waves consume from LDS after a workgroup barrier.
// ---------------------------------------------------------------------------
__global__ void __launch_bounds__(256) attn_part_kernel(
    const float* __restrict__ qkv, const float* __restrict__ kv_cache,
    const int* __restrict__ head_idx, float* __restrict__ ctx_p,
    float* __restrict__ ml, int ssplit)
{
    __shared__ float s_scores[S_MAX];                 // sized for ssplit==1 fallback
    __shared__ float s_q[D];
    __shared__ float s_red[256];
    __shared__ float s_tile[2][KT * ROWF];            // TDM double buffer (2x33KB)

    const int bj    = blockIdx.x;
    const int split = blockIdx.y;
    const int b = bj / TOPK;
    const int j = bj % TOPK;
    const int h = head_idx[b * TOPK + j] & (H - 1);
    for (int jj = 0; jj < j; ++jj)                    // dedupe (block-uniform)
        if ((head_idx[b * TOPK + jj] & (H - 1)) == h) return;

    const int chunk  = S_MAX / ssplit;
    const int s_lo   = split * chunk;
    const int ntiles = chunk / KT;                    // 8 (split) or 64 (fallback)
    const int tid    = threadIdx.x;
    const bool issuer = tid < 32;                     // wave 0 drives the TDM

    if (tid < D) s_q[tid] = qkv[(size_t)b * QKV_ROWS + h * D + tid];

    const float scale = 0.08838834764831845f;         // 1/sqrt(128)
    const float* knew = qkv + (size_t)b * QKV_ROWS + E     + h * D;
    const float* vnew = qkv + (size_t)b * QKV_ROWS + 2 * E + h * D;
    const float* kbase = kv_cache + (size_t)b * S_MAX * 2 * H * D + (size_t)h * D;
    const float* vbase = kbase + (size_t)H * D;
    const unsigned lds0 = (unsigned)(uintptr_t)(void*)&s_tile[0][0];
    const unsigned lds1 = (unsigned)(uintptr_t)(void*)&s_tile[1][0];

    auto valid_rows = [&](int s0) {                   // rows safe to read from cache
        int v = OFFSET - s0;                          // row s==OFFSET is stale
        return (v > KT) ? KT : v;
    };

    // ---------------- Phase 1: scores + chunk max (K tiles via TDM) --------
    if (issuer)
        tdm_load_2d(lds0, kbase + (size_t)s_lo * 2 * H * D, valid_rows(s_lo), KT);

    float lmax = -3.4e38f;
    for (int t = 0; t < ntiles; ++t) {
        const int s0 = s_lo + t * KT;
        if (issuer) {
            if (t + 1 < ntiles) {
                tdm_load_2d((t & 1) ? lds0 : lds1,
                            kbase + (size_t)(s0 + KT) * 2 * H * D,
                            valid_rows(s0 + KT), KT);
                __builtin_amdgcn_s_wait_tensorcnt(1); // tile t complete (in-order)
            } else {
                __builtin_amdgcn_s_wait_tensorcnt(0);
            }
        }
        __syncthreads();
        float* tile = &s_tile[t & 1][0];
        if (OFFSET >= s0 && OFFSET < s0 + KT && tid < D)
            tile[(OFFSET - s0) * ROWF + tid] = knew[tid];   // splice new K row
        __syncthreads();

        const int r = tid & (KT - 1);                 // tile row
        const int q = tid >> 6;                       // d-quarter
        float part = 0.f;
#pragma unroll
        for (int d = q * 32; d < q * 32 + 32; d += 4) {
            const float4 kv4 = *reinterpret_cast<const float4*>(tile + r * ROWF + d);
            part += kv4.x * s_q[d] + kv4.y * s_q[d + 1]
                  + kv4.z * s_q[d + 2] + kv4.w * s_q[d + 3];
        }
        s_red[tid] = part;
        __syncthreads();
        if (tid < KT) {
            const float sc = (s_red[tid] + s_red[tid + 64] +
                              s_red[tid + 128] + s_red[tid + 192]) * scale;
            s_scores[t * KT + tid] = sc;
            lmax = fmaxf(lmax, sc);
        }
        __syncthreads();
    }
    s_red[tid] = lmax;
    __syncthreads();
    for (int st = 128; st > 0; st >>= 1) {
        if (tid < st) s_red[tid] = fmaxf(s_red[tid], s_red[tid + st]);
        __syncthreads();
    }
    const float cmax = s_red[0];
    __syncthreads();

    // Prefetch V tile 0 now -> DMA overlaps the softmax phase
    if (issuer)
        tdm_load_2d(lds0, vbase + (size_t)s_lo * 2 * H * D, valid_rows(s_lo), KT);

    // ---------------- Phase 2: exponentials + chunk sum --------------------
    float lsum = 0.f;
    for (int s = s_lo + tid; s < s_lo + chunk; s += 256) {
        const float p = __expf(s_scores[s - s_lo] - cmax);
        s_scores[s - s_lo] = p;
        lsum += p;
    }
    s_red[tid] = lsum;
    __syncthreads();
    for (int st = 128; st > 0; st >>= 1) {
        if (tid < st) s_red[tid] += s_red[tid + st];
        __syncthreads();
    }
    const float csum = s_red[0];
    __syncthreads();

    // ---------------- Phase 3: ctx[d] = sum_s p[s]*v[s][d] (V via TDM) -----
    const int d     = tid & (D - 1);
    const int rhalf = tid >> 7;
    float acc = 0.f;
    for (int t = 0; t < ntiles; ++t) {
        const int s0 = s_lo + t * KT;
        if (issuer) {
            if (t + 1 < ntiles) {
                tdm_load_2d((t & 1) ? lds0 : lds1,
                            vbase + (size_t)(s0 + KT) * 2 * H * D,
                            valid_rows(s0 + KT), KT);
                __builtin_amdgcn_s_wait_tensorcnt(1);
            } else {
                __builtin_amdgcn_s_wait_tensorcnt(0);
            }
        }
        __syncthreads();
        float* tile = &s_tile[t & 1][0];
        if (OFFSET >= s0 && OFFSET < s0 + KT && tid < D)
            tile[(OFFSET - s0) * ROWF + tid] = vnew[tid];   // splice new V row
        __syncthreads();
#pragma unroll 8
        for (int r = rhalf; r < KT; r += 2)
            acc += s_scores[t * KT + r] * tile[r * ROWF + d];
        __syncthreads();
    }
    s_red[tid] = acc;
    __syncthreads();
    const int slot = bj * ssplit + split;
    if (rhalf == 0)
        ctx_p[(size_t)slot * D + d] = s_red[tid] + s_red[tid + 128];
    if (tid == 0) { ml[2 * slot] = cmax; ml[2 * slot + 1] = csum; }
}

// Merge s-chunk partials with log-sum-exp rescaling (fixed order).
__global__ void __launch_bounds__(128) attn_combine_kernel(
    const float* __restrict__ ctx_p, const float* __restrict__ ml,
    const int* __restrict__ head_idx, float* __restrict__ ctx, int ssplit)
{
    const int bj = blockIdx.x;
    const int b = bj / TOPK;
    const int j = bj % TOPK;
    const int h = head_idx[b * TOPK + j] & (H - 1);
    for (int jj = 0; jj < j; ++jj)
        if ((head_idx[b * TOPK + jj] & (H - 1)) == h) return;

    const int d = threadIdx.x;
    float M = -3.4e38f;
    for (int i = 0; i < ssplit; ++i)
        M = fmaxf(M, ml[2 * (bj * ssplit + i)]);
    float l = 0.f, c = 0.f;
    for (int i = 0; i < ssplit; ++i) {
        const int slot = bj * ssplit + i;
        const float w = __expf(ml[2 * slot] - M);
        l += w * ml[2 * slot + 1];
        c += w * ctx_p[(size_t)slot * D + d];
    }
    ctx[(size_t)b * E + h * D + d] = c / l;
}

// ---------------------------------------------------------------------------
extern "C" void kernel_launch(void* const* d_in, const int* in_sizes, int n_in,
                              void* d_out, int out_size, void* d_ws, size_t ws_size,
                              hipStream_t stream)
{
    const float* x        = (const float*)d_in[0];   // (B,1,E)
    const float* kv_cache = (const float*)d_in[1];   // (B,S_MAX,2,H,D)
    const float* w_qkv    = (const float*)d_in[2];   // (3E,E)
    const float* b_qkv    = (const float*)d_in[3];   // (3E)
    const float* w_out    = (const float*)d_in[4];   // (E,E)
    const float* b_out    = (const float*)d_in[5];   // (E)
    const int*   head_idx = (const int*)d_in[6];     // (B,TOPK)
    float*       out      = (float*)d_out;           // (B,1,E)

    int nsA = 2, nsC = 4, ssplit = 8;
    {
        const size_t full_floats =
            (size_t)B * QKV_ROWS + (size_t)B * E +
            (size_t)B * TOPK * 8 * D + (size_t)B * TOPK * 8 * 2 +
            (size_t)2 * B * QKV_ROWS + (size_t)4 * B * E;
        if (ws_size < full_floats * sizeof(float)) { nsA = 1; nsC = 1; ssplit = 1; }
    }

    size_t off = 0;
    auto take = [&](size_t nfloats) {
        float* p = (float*)d_ws + off; off += nfloats; return p;
    };
    float* qkv_ws = take((size_t)B * QKV_ROWS);
    float* ctx_ws = take((size_t)B * E);
    float* actx_p = take((size_t)B * TOPK * ssplit * D);
    float* aml    = take((size_t)B * TOPK * ssplit * 2);
    float* partA  = take((size_t)nsA * B * QKV_ROWS);
    float* partC  = take((size_t)nsC * B * E);

    hipMemsetAsync(ctx_ws, 0, (size_t)B * E * sizeof(float), stream);

    gemv_part_kernel<true><<<dim3(QKV_ROWS / 16 / 8, nsA), 256, 0, stream>>>(
        w_qkv, x, head_idx, partA, QKV_ROWS, nsA);
    gemv_combine_kernel<<<(B * QKV_ROWS + 255) / 256, 256, 0, stream>>>(
        partA, b_qkv, qkv_ws, QKV_ROWS, nsA);

    attn_part_kernel<<<dim3(B * TOPK, ssplit), 256, 0, stream>>>(
        qkv_ws, kv_cache, head_idx, actx_p, aml, ssplit);
    attn_combine_kernel<<<B * TOPK, 128, 0, stream>>>(
        actx_p, aml, head_idx, ctx_ws, ssplit);

    gemv_part_kernel<false><<<dim3(E / 16 / 8, nsC), 256, 0, stream>>>(
        w_out, ctx_ws, head_idx, partC, E, nsC);
    gemv_combine_kernel<<<(B * E + 255) / 256, 256, 0, stream>>>(
        partC, b_out, out, E, nsC);
}